// MatrixVectorScaledDotProductAttention_91345364451422
// MI455X (gfx1250) — compile-verified
//
#include <hip/hip_runtime.h>

// MI455X / gfx1250, wave32. Bandwidth-bound batched attention:
//   scores  = (q . K^T)/8   -> V_WMMA_F32_16X16X4_F32 (q replicated across B columns)
//   softmax = LDS-resident row (2048 floats) with tree reductions
//   out     = attn . V      -> coalesced float2 streaming accumulation

typedef float v2f __attribute__((ext_vector_type(2)));
typedef float v8f __attribute__((ext_vector_type(8)));

#define DK       64
#define DV       64
#define LSEQ     2048
#define NTHREADS 256
#define NWAVES   8

__launch_bounds__(NTHREADS)
__global__ void mvsdpa_kernel(const float* __restrict__ q,
                              const float* __restrict__ k,
                              const float* __restrict__ v,
                              float* __restrict__ out,
                              float* __restrict__ attn)
{
    __shared__ float s_smem[LSEQ];     // scores -> exp -> normalized attn
    __shared__ float red[NTHREADS];    // block reductions
    __shared__ float q_smem[DK];
    __shared__ float out_smem[DV];

    const int b    = blockIdx.x;
    const int tid  = threadIdx.x;
    const int wave = tid >> 5;
    const int lane = tid & 31;

    const float* kb = k + (size_t)b * LSEQ * DK;
    const float* vb = v + (size_t)b * LSEQ * DV;

    if (tid < DK) q_smem[tid]  = q[b * DK + tid];
    if (tid < DV) out_smem[tid] = 0.0f;
    __syncthreads();

    // ---------------- Phase 1: scores via WMMA f32 16x16x4 ----------------
    // A (16x4 f32 layout): lane L holds row M = L%16; VGPR0 = K=(L<16?0:2),
    // VGPR1 = K+1.  B (4x16): row k striped the same way; we replicate q[k]
    // across all 16 columns, so every column of D is the score vector.
    const int m    = lane & 15;
    const int koff = (lane < 16) ? 0 : 2;

    for (int tile = wave; tile < LSEQ / 16; tile += NWAVES) {
        const int l_base = tile * 16;
        const float* arow = kb + (size_t)(l_base + m) * DK + koff;
        v8f acc = {};
        #pragma unroll
        for (int d = 0; d < DK; d += 4) {
            v2f a;
            a.x = arow[d];         // k[b, l_base+m, d+koff]
            a.y = arow[d + 1];     // k[b, l_base+m, d+koff+1]
            v2f bm;
            bm.x = q_smem[d + koff];
            bm.y = q_smem[d + koff + 1];
            acc = __builtin_amdgcn_wmma_f32_16x16x4_f32(
                false, a, false, bm, (short)0, acc, false, false);
        }
        // C/D layout: VGPR j -> M=j (lanes 0-15), M=j+8 (lanes 16-31).
        if (lane == 0) {
            #pragma unroll
            for (int j = 0; j < 8; ++j)
                s_smem[l_base + j] = acc[j] * 0.125f;          // /TEMPERATURE
        }
        if (lane == 16) {
            #pragma unroll
            for (int j = 0; j < 8; ++j)
                s_smem[l_base + 8 + j] = acc[j] * 0.125f;
        }
    }
    __syncthreads();

    // ---------------- Phase 2: softmax over the 2048-score row -------------
    float lmax = -3.4e38f;
    #pragma unroll
    for (int i = 0; i < LSEQ / NTHREADS; ++i)
        lmax = fmaxf(lmax, s_smem[tid + i * NTHREADS]);
    red[tid] = lmax;
    __syncthreads();
    for (int off = NTHREADS / 2; off > 0; off >>= 1) {
        if (tid < off) red[tid] = fmaxf(red[tid], red[tid + off]);
        __syncthreads();
    }
    const float mx = red[0];
    __syncthreads();

    float lsum = 0.0f;
    #pragma unroll
    for (int i = 0; i < LSEQ / NTHREADS; ++i) {
        const int idx = tid + i * NTHREADS;
        const float e = __expf(s_smem[idx] - mx);
        s_smem[idx] = e;
        lsum += e;
    }
    red[tid] = lsum;
    __syncthreads();
    for (int off = NTHREADS / 2; off > 0; off >>= 1) {
        if (tid < off) red[tid] += red[tid + off];
        __syncthreads();
    }
    const float invZ = 1.0f / red[0];

    // Normalize in LDS and stream attn out (coalesced).
    float* attn_b = attn + (size_t)b * LSEQ;
    #pragma unroll
    for (int i = 0; i < LSEQ / NTHREADS; ++i) {
        const int idx = tid + i * NTHREADS;
        const float a = s_smem[idx] * invZ;
        s_smem[idx] = a;
        attn_b[idx] = a;
    }
    __syncthreads();

    // ---------------- Phase 3: out = attn . V (streaming, coalesced) ------
    // Wave w owns l in [w*256, w*256+256); lane accumulates e=2*lane, 2*lane+1.
    const float2* vb2 = (const float2*)vb;
    float accx = 0.0f, accy = 0.0f;
    const int l_lo = wave * (LSEQ / NWAVES);
    #pragma unroll 4
    for (int l = l_lo; l < l_lo + (LSEQ / NWAVES); ++l) {
        const float  a  = s_smem[l];                 // LDS broadcast
        const float2 vv = vb2[l * (DV / 2) + lane];  // 256B/wave, b64 loads
        accx += a * vv.x;
        accy += a * vv.y;
    }
    atomicAdd(&out_smem[2 * lane],     accx);        // ds_add_f32
    atomicAdd(&out_smem[2 * lane + 1], accy);
    __syncthreads();

    if (tid < DV) out[b * DV + tid] = out_smem[tid];
}

extern "C" void kernel_launch(void* const* d_in, const int* in_sizes, int n_in,
                              void* d_out, int out_size, void* d_ws, size_t ws_size,
                              hipStream_t stream) {
    const float* q = (const float*)d_in[0];
    const float* k = (const float*)d_in[1];
    const float* v = (const float*)d_in[2];
    const int NB = in_sizes[0] / DK;                 // 2048

    float* out  = (float*)d_out;                     // (NB, DV) first
    float* attn = out + (size_t)NB * DV;             // then (NB, LSEQ)

    mvsdpa_kernel<<<dim3(NB), dim3(NTHREADS), 0, stream>>>(q, k, v, out, attn);
}